// SetCriterion_23914377904891
// MI455X (gfx1250) — compile-verified
//
#include <hip/hip_runtime.h>
#include <hip/hip_bf16.h>
#include <math.h>

// ---------------- problem constants (from reference setup_inputs) -----------
#define BB 16          // batch
#define CC 12          // classes
#define HH 160
#define WW 160
#define NN 100         // boxes per batch
#define ALPHA 0.75f

#define CHUNKS 8                   // row chunks per (b,c) plane
#define ROWS   (HH / CHUNKS)       // 20 rows per block
#define W4     (WW / 4)            // 40 float4 per row
#define NBLK   (BB * CC * CHUNKS)  // 1536 blocks

typedef __attribute__((ext_vector_type(2))) float v2f;
typedef __attribute__((ext_vector_type(4))) float f4;
typedef __attribute__((ext_vector_type(8))) float v8f;

// Numerically-stable sigmoid focal term for one element (forward value only).
__device__ __forceinline__ float focal_term(float x, float t) {
    float e    = __expf(-fabsf(x));                       // exp(-|x|)
    float ce   = fmaxf(x, 0.f) - x * t + log1pf(e);       // BCE with logits
    float inv  = 1.f / (1.f + e);
    float prob = (x >= 0.f) ? inv : e * inv;              // sigmoid(x)
    float p_t  = prob * t + (1.f - prob) * (1.f - t);
    float om   = 1.f - p_t;
    float a_t  = 0.25f + 0.5f * t;                        // ALPHA*t+(1-ALPHA)*(1-t)
    return a_t * ce * om * om;                            // gamma = 2
}

// Exact f32 wave32 sum via V_WMMA_F32_16X16X4_F32:
// A row0 = all ones (lanes 0 and 16 set both A-VGPRs), other rows 0.
// B = (val, 0): whichever K rows B's VGPR0 maps to, D[0][n] = val[n]+val[n+16].
// Result lives in d[0], lanes 0..15; finish with 4 xor-shuffles.
__device__ __forceinline__ float wave_sum_wmma(float val, int lane) {
    int m = lane & 15;
    v2f a; a.x = (m == 0) ? 1.f : 0.f; a.y = a.x;
    v2f b; b.x = val; b.y = 0.f;
    v8f c = {};
    v8f d = __builtin_amdgcn_wmma_f32_16x16x4_f32(
        /*neg_a=*/false, a, /*neg_b=*/false, b,
        /*c_mod=*/(short)0, c, /*reuse_a=*/false, /*reuse_b=*/false);
    float r = d[0];                       // lane n (0..15): val[n]+val[n+16]
    r += __shfl_xor(r, 8);
    r += __shfl_xor(r, 4);
    r += __shfl_xor(r, 2);
    r += __shfl_xor(r, 1);
    return r;                             // lanes 0..15 hold the sum; lanes 16..31 hold 0
}

// One block per (batch, class, row-chunk). Builds the relevant box list in
// LDS, streams aux_pred once (nontemporal f4 loads), computes the on-the-fly
// heatmap max + focal term, reduces with WMMA, writes one partial pair.
__global__ void __launch_bounds__(256)
focal_fused_kernel(const float* __restrict__ aux,
                   const float* __restrict__ boxes,
                   const int*   __restrict__ labels,
                   float*       __restrict__ partials) {
    __shared__ float4 sbox[NN];
    __shared__ int    scount;
    __shared__ float  sred[16];

    const int tid   = threadIdx.x;
    const int plane = blockIdx.x >> 3;        // / CHUNKS
    const int chunk = blockIdx.x & (CHUNKS - 1);
    const int b     = plane / CC;
    const int c     = plane - b * CC;
    const int y0    = chunk * ROWS;
    const int y1    = y0 + ROWS - 1;

    if (tid == 0) scount = 0;
    __syncthreads();

    // Build the per-(b,c) box list, filtered to this row chunk.
    for (int n = tid; n < NN; n += blockDim.x) {
        if (labels[b * NN + n] == c) {
            const float* bp = boxes + (size_t)(b * NN + n) * 5;
            float cx = bp[0], cy = bp[1], w = bp[2], h = bp[3];
            float ix = fminf(fmaxf(floorf(cx * (float)WW), 0.f), (float)(WW - 1));
            float iy = fminf(fmaxf(floorf(cy * (float)HH), 0.f), (float)(HH - 1));
            float base = sqrtf(w * (float)WW * h * (float)HH);
            float r = fmaxf(floorf(2.f * log1pf(base)), 1.f);
            if (iy - r <= (float)y1 && iy + r >= (float)y0) {
                float sig  = (2.f * r + 1.f) * (1.f / 6.f);
                float inv2 = 1.f / (2.f * sig * sig);
                int slot = atomicAdd(&scount, 1);
                sbox[slot] = make_float4(ix, iy, r, inv2);
            }
        }
    }
    __syncthreads();
    const int nb = scount;

    const f4* pred4 = (const f4*)(aux + (size_t)plane * HH * WW);
    float s_term = 0.f, s_t = 0.f;

    for (int v = tid; v < ROWS * W4; v += blockDim.x) {
        int yy = y0 + v / W4;
        int xc = v - (v / W4) * W4;
        f4 p = __builtin_nontemporal_load(&pred4[(size_t)yy * W4 + xc]);

        float yf = (float)yy;
        float x0 = (float)(xc * 4);
        float t0 = 0.f, t1 = 0.f, t2 = 0.f, t3 = 0.f;
        for (int j = 0; j < nb; ++j) {
            float4 bx = sbox[j];
            float dy = yf - bx.y;
            if (fabsf(dy) <= bx.z) {
                float dy2 = dy * dy;
                float dx = x0 - bx.x;
                if (fabsf(dx) <= bx.z) t0 = fmaxf(t0, __expf(-(dx * dx + dy2) * bx.w));
                dx += 1.f;
                if (fabsf(dx) <= bx.z) t1 = fmaxf(t1, __expf(-(dx * dx + dy2) * bx.w));
                dx += 1.f;
                if (fabsf(dx) <= bx.z) t2 = fmaxf(t2, __expf(-(dx * dx + dy2) * bx.w));
                dx += 1.f;
                if (fabsf(dx) <= bx.z) t3 = fmaxf(t3, __expf(-(dx * dx + dy2) * bx.w));
            }
        }
        s_term += focal_term(p.x, t0) + focal_term(p.y, t1)
                + focal_term(p.z, t2) + focal_term(p.w, t3);
        s_t    += t0 + t1 + t2 + t3;
    }

    // ---- wave32 reduction via V_WMMA_F32_16X16X4_F32 (exact f32) ----
    const int lane = tid & 31;
    const int wave = tid >> 5;
    float wt = wave_sum_wmma(s_term, lane);
    float wb = wave_sum_wmma(s_t, lane);
    if (lane == 0) { sred[wave * 2] = wt; sred[wave * 2 + 1] = wb; }
    __syncthreads();

    if (tid == 0) {
        float a = 0.f, bsum = 0.f;
        #pragma unroll
        for (int i = 0; i < 8; ++i) { a += sred[2 * i]; bsum += sred[2 * i + 1]; }
        partials[(size_t)blockIdx.x * 2 + 0] = a;
        partials[(size_t)blockIdx.x * 2 + 1] = bsum;
    }
}

// Deterministic fixed-order final reduction of the 1536 partial pairs.
__global__ void __launch_bounds__(256)
focal_finalize_kernel(const float* __restrict__ partials,
                      float* __restrict__ out, int nblk) {
    __shared__ float st[256];
    __shared__ float sb[256];
    float a = 0.f, b = 0.f;
    for (int i = threadIdx.x; i < nblk; i += 256) {
        a += partials[2 * i];
        b += partials[2 * i + 1];
    }
    st[threadIdx.x] = a;
    sb[threadIdx.x] = b;
    __syncthreads();
    for (int s = 128; s > 0; s >>= 1) {
        if (threadIdx.x < s) {
            st[threadIdx.x] += st[threadIdx.x + s];
            sb[threadIdx.x] += sb[threadIdx.x + s];
        }
        __syncthreads();
    }
    if (threadIdx.x == 0) out[0] = st[0] / fmaxf(sb[0], 1.f);
}

extern "C" void kernel_launch(void* const* d_in, const int* in_sizes, int n_in,
                              void* d_out, int out_size, void* d_ws, size_t ws_size,
                              hipStream_t stream) {
    const float* aux    = (const float*)d_in[0];  // [16,12,160,160] f32
    const float* boxes  = (const float*)d_in[1];  // [16,100,5] f32
    const int*   labels = (const int*)d_in[2];    // [16,100] i32
    float* out  = (float*)d_out;                  // scalar f32
    float* part = (float*)d_ws;                   // 1536*2 floats = 12 KB

    focal_fused_kernel<<<NBLK, 256, 0, stream>>>(aux, boxes, labels, part);
    focal_finalize_kernel<<<1, 256, 0, stream>>>(part, out, NBLK);
}